// LieTransport_49392123904695
// MI455X (gfx1250) — compile-verified
//
#include <hip/hip_runtime.h>

// LieTransport: flow-warped bilinear resampling of h_prev[B,C,H,W,R].
// Memory-bound (≈0.7 FLOP/byte → ~33us HBM roofline on MI455X @ 23.3 TB/s).
// Strategy: one thread per (b,c,y,x); 16x global_load_b128 corner gathers
// (64B-contiguous R runs, fully consumed), non-temporal b128 stores to keep
// the write-once output out of L2 so L2 serves gather-read reuse.

typedef __attribute__((ext_vector_type(4))) float v4f;

__global__ __launch_bounds__(128) void lie_transport_kernel(
    const float* __restrict__ h_prev,   // [B,C,H,W,R]
    const float* __restrict__ flow,     // [B,2,H,W]
    const float* __restrict__ dt,       // [B]
    float*       __restrict__ out)      // [B,C,H,W,R]
{
    constexpr int B = 4, C = 64, H = 128, W = 128, R = 16;
    constexpr int RV = R / 4;                   // float4 chunks per pixel-channel
    constexpr float STEPX = 2.0f / (float)(W - 1);
    constexpr float STEPY = 2.0f / (float)(H - 1);

    const int x = threadIdx.x;                  // 0..127 (== W)
    const int y = blockIdx.x;                   // 0..H-1
    const int c = blockIdx.y;                   // 0..C-1
    const int b = blockIdx.z;                   // 0..B-1

    // ---- per-pixel sampling coordinates (reference math, div-free) ----
    const float dtb = dt[b];                    // uniform -> scalar load
    const float fx  = flow[((size_t)(b * 2 + 0) * H + y) * W + x];
    const float fy  = flow[((size_t)(b * 2 + 1) * H + y) * W + x];

    const float base_x = __builtin_fmaf((float)x, STEPX, -1.0f);
    const float base_y = __builtin_fmaf((float)y, STEPY, -1.0f);
    const float gx = base_x - fx * dtb;
    const float gy = base_y - fy * dtb;

    // ((g+1)*N - 1) * 0.5  ==  g*(N/2) + (N-1)/2
    float ix = __builtin_fmaf(gx, (float)W * 0.5f, (float)(W - 1) * 0.5f);
    float iy = __builtin_fmaf(gy, (float)H * 0.5f, (float)(H - 1) * 0.5f);
    ix = fminf(fmaxf(ix, 0.0f), (float)(W - 1));
    iy = fminf(fmaxf(iy, 0.0f), (float)(H - 1));

    const float x0f = floorf(ix);
    const float y0f = floorf(iy);
    const float wx  = ix - x0f;
    const float wy  = iy - y0f;

    int x0i = (int)x0f;  x0i = x0i < 0 ? 0 : (x0i > W - 1 ? W - 1 : x0i);
    int y0i = (int)y0f;  y0i = y0i < 0 ? 0 : (y0i > H - 1 ? H - 1 : y0i);
    const int x1i = x0i + 1 > W - 1 ? W - 1 : x0i + 1;
    const int y1i = y0i + 1 > H - 1 ? H - 1 : y0i + 1;

    const float omwx = 1.0f - wx;
    const float omwy = 1.0f - wy;

    // ---- pointers: every corner is a 64B-contiguous run over R ----
    const size_t planeOff = (size_t)((b * C + c) * H) * (W * R);
    const float* plane = h_prev + planeOff;

    const v4f* __restrict__ p00 = reinterpret_cast<const v4f*>(plane + (size_t)(y0i * W + x0i) * R);
    const v4f* __restrict__ p01 = reinterpret_cast<const v4f*>(plane + (size_t)(y0i * W + x1i) * R);
    const v4f* __restrict__ p10 = reinterpret_cast<const v4f*>(plane + (size_t)(y1i * W + x0i) * R);
    const v4f* __restrict__ p11 = reinterpret_cast<const v4f*>(plane + (size_t)(y1i * W + x1i) * R);
    v4f*       __restrict__ po  = reinterpret_cast<v4f*>(out + planeOff + (size_t)(y * W + x) * R);

#pragma unroll
    for (int j = 0; j < RV; ++j) {
        const v4f v00 = p00[j];   // global_load_b128
        const v4f v01 = p01[j];
        const v4f v10 = p10[j];
        const v4f v11 = p11[j];

        const v4f top = v00 * omwx + v01 * wx;
        const v4f bot = v10 * omwx + v11 * wx;
        const v4f r   = top * omwy + bot * wy;

        // write-once output: non-temporal (gfx1250 TH=NT) b128 store
        __builtin_nontemporal_store(r, po + j);
    }
}

extern "C" void kernel_launch(void* const* d_in, const int* in_sizes, int n_in,
                              void* d_out, int out_size, void* d_ws, size_t ws_size,
                              hipStream_t stream) {
    (void)in_sizes; (void)n_in; (void)d_ws; (void)ws_size; (void)out_size;

    const float* h_prev = (const float*)d_in[0];   // [4,64,128,128,16] fp32
    const float* flow   = (const float*)d_in[1];   // [4,2,128,128]     fp32
    const float* dt     = (const float*)d_in[2];   // [4]               fp32
    float*       out    = (float*)d_out;

    constexpr int B = 4, C = 64, H = 128;
    dim3 grid(H, C, B);                            // y, c, b  (32768 blocks)
    lie_transport_kernel<<<grid, 128, 0, stream>>>(h_prev, flow, dt, out);
}